// Input_Net_29317446762762
// MI455X (gfx1250) — compile-verified
//
#include <hip/hip_runtime.h>
#include <stdint.h>

#define NTGT    8192
#define NSRC    16384
#define NBATCH  8
#define NH      10
#define THREADS 256
#define TILE    2048
#define NTILES  (NSRC / TILE)
#define PT      (TILE / THREADS)   /* 8 floats per thread per tile per array */

#define OFF_XNEAR 0
#define OFF_INTER (NBATCH * NTGT * NH)          /* 655360 */
#define OFF_DIST  (OFF_INTER + NBATCH * NTGT)   /* 720896 */
#define OFF_LON   (OFF_DIST + NTGT * NH)        /* 802816 */
#define OFF_LAT   (OFF_LON + NTGT * NH)         /* 884736 */

// CDNA5 async copy: global -> LDS, tracked by ASYNCcnt (CDNA5 ISA).
// VDST = VGPR holding LDS byte address, VADDR = 64-bit global address, SADDR = off.
__device__ __forceinline__ void async_ld_b128(uint32_t lds_off, uint64_t gaddr) {
  asm volatile("global_load_async_to_lds_b128 %0, %1, off"
               :: "v"(lds_off), "v"(gaddr) : "memory");
}
__device__ __forceinline__ void wait_async0() {
  asm volatile("s_wait_asynccnt 0" ::: "memory");
}

__global__ __launch_bounds__(THREADS)
void knn_topk_interp_kernel(const float* __restrict__ x,
                            const float* __restrict__ dlon,
                            const float* __restrict__ dlat,
                            float* __restrict__ out)
{
  __shared__ float  s_lon[2][TILE];
  __shared__ float  s_lat[2][TILE];
  __shared__ double s_wmin[THREADS / 32];
  __shared__ double s_res[NH];

  const int tid = threadIdx.x;
  const int row = blockIdx.x;
  const uint64_t g_lon_row = (uint64_t)(uintptr_t)(dlon + (size_t)row * NSRC);
  const uint64_t g_lat_row = (uint64_t)(uintptr_t)(dlat + (size_t)row * NSRC);

  const double DINF = __builtin_inf();
  double L[NH];
#pragma unroll
  for (int i = 0; i < NH; ++i) L[i] = DINF;

  // LDS byte addresses: flat shared-pointer low 32 bits == LDS address (ISA 10.2).
  const uint32_t l_lon0 = (uint32_t)(uintptr_t)&s_lon[0][tid * PT];
  const uint32_t l_lat0 = (uint32_t)(uintptr_t)&s_lat[0][tid * PT];
  const uint32_t bufstride = (uint32_t)(TILE * sizeof(float));

  // Prologue: issue tile 0 into buffer 0.
  {
    uint64_t gl = g_lon_row + (uint64_t)(tid * PT) * 4u;
    uint64_t ga = g_lat_row + (uint64_t)(tid * PT) * 4u;
    async_ld_b128(l_lon0,      gl);
    async_ld_b128(l_lon0 + 16, gl + 16);
    async_ld_b128(l_lat0,      ga);
    async_ld_b128(l_lat0 + 16, ga + 16);
  }
  wait_async0();
  __syncthreads();

  for (int t = 0; t < NTILES; ++t) {
    const int cur = t & 1;
    if (t + 1 < NTILES) {
      const int nxt = cur ^ 1;
      uint64_t gl = g_lon_row + (uint64_t)((t + 1) * TILE + tid * PT) * 4u;
      uint64_t ga = g_lat_row + (uint64_t)((t + 1) * TILE + tid * PT) * 4u;
      uint32_t ll = l_lon0 + (uint32_t)nxt * bufstride;
      uint32_t la = l_lat0 + (uint32_t)nxt * bufstride;
      async_ld_b128(ll,      gl);
      async_ld_b128(ll + 16, gl + 16);
      async_ld_b128(la,      ga);
      async_ld_b128(la + 16, ga + 16);
    }

    // Consume current tile from LDS (ds_load_b128 x4), overlapping next tile's async loads.
    const int base = t * TILE + tid * PT;
    const float4 lo0 = *(const float4*)&s_lon[cur][tid * PT];
    const float4 lo1 = *(const float4*)&s_lon[cur][tid * PT + 4];
    const float4 la0 = *(const float4*)&s_lat[cur][tid * PT];
    const float4 la1 = *(const float4*)&s_lat[cur][tid * PT + 4];
    float lov[PT] = {lo0.x, lo0.y, lo0.z, lo0.w, lo1.x, lo1.y, lo1.z, lo1.w};
    float lav[PT] = {la0.x, la0.y, la0.z, la0.w, la1.x, la1.y, la1.z, la1.w};
#pragma unroll
    for (int j = 0; j < PT; ++j) {
      const float lo = lov[j], la = lav[j];
      const float r2 = __fadd_rn(__fmul_rn(lo, lo), __fmul_rn(la, la)); // match XLA: no FMA
      const float d  = sqrtf(r2);
      // Exact (dist, idx) lexicographic key as a positive double.
      const unsigned long long kb =
          __double_as_longlong((double)d) | (unsigned long long)(base + j);
      const double key = __longlong_as_double(kb);
      if (key < L[NH - 1]) {               // sorted-insert, drop current max
        double c = key;
#pragma unroll
        for (int q = 0; q < NH; ++q) {
          const double mn = fmin(c, L[q]);
          const double mx = fmax(c, L[q]);
          L[q] = mn;
          c = mx;
        }
      }
    }
    wait_async0();      // tile t+1 data landed (per-wave ASYNCcnt)
    __syncthreads();    // visible to all waves; buffer reuse safe
  }

  // Merge 256 sorted lists: 10 rounds of workgroup-min over list heads.
  const int lane = tid & 31;
  const int wid  = tid >> 5;
  for (int r = 0; r < NH; ++r) {
    double k = L[0];
#pragma unroll
    for (int off = 16; off >= 1; off >>= 1)
      k = fmin(k, __shfl_xor(k, off, 32));
    __syncthreads();
    if (lane == 0) s_wmin[wid] = k;
    __syncthreads();
    double m = s_wmin[0];
#pragma unroll
    for (int w = 1; w < THREADS / 32; ++w) m = fmin(m, s_wmin[w]);
    if (L[0] == m) {                       // unique winner (idx embedded in key)
      s_res[r] = m;
#pragma unroll
      for (int q = 0; q < NH - 1; ++q) L[q] = L[q + 1];
      L[NH - 1] = DINF;
    }
  }
  __syncthreads();

  // ---- Outputs ----
  if (tid < NH) {
    const unsigned long long kb = __double_as_longlong(s_res[tid]);
    const int   idx = (int)(kb & 0x1FFFFFFFULL);
    const float d   = (float)__longlong_as_double(kb & ~0x1FFFFFFFULL);
    out[OFF_DIST + (size_t)row * NH + tid] = d;
    out[OFF_LON  + (size_t)row * NH + tid] = dlon[(size_t)row * NSRC + idx];
    out[OFF_LAT  + (size_t)row * NH + tid] = dlat[(size_t)row * NSRC + idx];
  }
  if (tid < NBATCH * NH) {
    const int b = tid / NH, n = tid % NH;
    const unsigned long long kb = __double_as_longlong(s_res[n]);
    const int idx = (int)(kb & 0x1FFFFFFFULL);
    out[OFF_XNEAR + ((size_t)b * NTGT + row) * NH + n] = x[(size_t)b * NSRC + idx];
  }
  if (tid < NBATCH) {
    const int b = tid;
    float num = 0.0f, den = 0.0f;
#pragma unroll
    for (int n = 0; n < NH; ++n) {
      const unsigned long long kb = __double_as_longlong(s_res[n]);
      const int   idx = (int)(kb & 0x1FFFFFFFULL);
      const float d   = (float)__longlong_as_double(kb & ~0x1FFFFFFFULL);
      const float w   = 1.0f / (d + 1e-10f);
      den += w;
      num += w * x[(size_t)b * NSRC + idx];
    }
    out[OFF_INTER + (size_t)b * NTGT + row] = num / den;
  }
}

extern "C" void kernel_launch(void* const* d_in, const int* in_sizes, int n_in,
                              void* d_out, int out_size, void* d_ws, size_t ws_size,
                              hipStream_t stream) {
  (void)in_sizes; (void)n_in; (void)d_ws; (void)ws_size; (void)out_size;
  const float* x    = (const float*)d_in[0];
  const float* dlon = (const float*)d_in[1];
  const float* dlat = (const float*)d_in[2];
  float* out = (float*)d_out;
  knn_topk_interp_kernel<<<NTGT, THREADS, 0, stream>>>(x, dlon, dlat, out);
}